// GraphMultiheadAttention_65197603553464
// MI455X (gfx1250) — compile-verified
//
#include <hip/hip_runtime.h>

#define N_NODES 50000
#define N_EDGES 800000
#define DIM     128
#define NHEAD   8
#define HEADDIM 16
#define INV_SQRT_HD 0.25f    // 1/sqrt(16)
#define LDS_PITCH 132        // 128 + 4 floats padding (keeps 16B align, breaks bank conflicts)

typedef float v2f __attribute__((ext_vector_type(2)));
typedef float v4f __attribute__((ext_vector_type(4)));
typedef float v8f __attribute__((ext_vector_type(8)));

// ---- monotone uint encoding of f32 so atomicMax(uint) == float max ----
__device__ __forceinline__ unsigned f32_ord(float f) {
    unsigned u = __float_as_uint(f);
    return (u & 0x80000000u) ? ~u : (u | 0x80000000u);
}
__device__ __forceinline__ float ord_f32(unsigned u) {
    return __uint_as_float((u & 0x80000000u) ? (u & 0x7fffffffu) : ~u);
}
#define ORD_NEG_INF 0x007FFFFFu   // f32_ord(-inf)

// -----------------------------------------------------------------------
// Kernel 0: init segment-max / segment-sum accumulators
// -----------------------------------------------------------------------
__global__ void init_kernel(unsigned* __restrict__ amax, float* __restrict__ denom) {
    int i = blockIdx.x * 256 + threadIdx.x;
    if (i < N_NODES * NHEAD) {
        amax[i]  = ORD_NEG_INF;
        denom[i] = 0.0f;
    }
}

// -----------------------------------------------------------------------
// Kernel 1: node projections Q,K,V (workspace) + OUT = h@W_h0 + b_h0
// 128 rows per block, 8 waves, each wave owns a 16-col strip.
// -----------------------------------------------------------------------
__global__ void node_proj_kernel(const float* __restrict__ h,
                                 const float* __restrict__ Wq, const float* __restrict__ bq,
                                 const float* __restrict__ Wk, const float* __restrict__ bk,
                                 const float* __restrict__ Wv, const float* __restrict__ bv,
                                 const float* __restrict__ Wh0, const float* __restrict__ bh0,
                                 float* __restrict__ Q, float* __restrict__ K,
                                 float* __restrict__ V, float* __restrict__ OUT) {
    extern __shared__ float lds[];                 // Xs[128][LDS_PITCH]
    const int tid = threadIdx.x, lane = tid & 31, wave = tid >> 5;
    const int r0 = blockIdx.x * 128;

    // Stage 128x128 f32 tile of h into LDS (coalesced float4 loads).
#pragma unroll
    for (int i = 0; i < 16; ++i) {
        int f4 = tid + 256 * i;                    // 4096 float4s
        int row = f4 >> 5, c4 = f4 & 31;
        v4f val = {};
        if (r0 + row < N_NODES)
            val = *(const v4f*)(h + (size_t)(r0 + row) * DIM + c4 * 4);
        *(v4f*)(lds + row * LDS_PITCH + c4 * 4) = val;
    }
    __syncthreads();

    const float* Ws[4] = {Wq, Wk, Wv, Wh0};
    const float* bs[4] = {bq, bk, bv, bh0};
    float*       Ys[4] = {Q, K, V, OUT};

    const int col   = (lane & 15) + wave * 16;     // output column of this lane
    const int koff  = (lane >> 4) * 2;             // K sub-offset (0 or 2)
    const int arow0 = lane & 15;                   // A-matrix row within tile

#pragma unroll 1
    for (int w = 0; w < 4; ++w) {
        const float* W = Ws[w];
        float* Y = Ys[w];
        v2f bfrag[32];
#pragma unroll
        for (int kt = 0; kt < 32; ++kt) {          // preload 128x16 strip of W
            int k = kt * 4 + koff;
            bfrag[kt].x = W[(size_t)k * DIM + col];
            bfrag[kt].y = W[(size_t)(k + 1) * DIM + col];
        }
        float bias = bs[w][col];
#pragma unroll 1
        for (int m = 0; m < 8; ++m) {              // 8 row sub-tiles of 16
            const float* xrow = lds + (m * 16 + arow0) * LDS_PITCH + koff;
            v8f acc = {};
#pragma unroll
            for (int kt = 0; kt < 32; ++kt) {
                v2f a = *(const v2f*)(xrow + kt * 4);
                acc = __builtin_amdgcn_wmma_f32_16x16x4_f32(
                        false, a, false, bfrag[kt], (short)0, acc, false, false);
            }
            int mbase = r0 + m * 16 + (lane >> 4) * 8;
#pragma unroll
            for (int j = 0; j < 8; ++j) {
                int grow = mbase + j;
                if (grow < N_NODES)
                    Y[(size_t)grow * DIM + col] = acc[j] + bias;
            }
        }
    }
}

// -----------------------------------------------------------------------
// Shared GEMM stage for edge kernels: x_ij = e_ij_tile @ W_e + b_e -> Xe (LDS)
// -----------------------------------------------------------------------
__device__ __forceinline__ void edge_tile_gemm(const float* __restrict__ e_ij,
                                               const float* __restrict__ We,
                                               const float* __restrict__ be,
                                               float* Xs, float* Xe, int e0) {
    const int tid = threadIdx.x, lane = tid & 31, wave = tid >> 5;
#pragma unroll
    for (int i = 0; i < 16; ++i) {
        int f4 = tid + 256 * i;
        int row = f4 >> 5, c4 = f4 & 31;
        const float* gp = e_ij + (size_t)(e0 + row) * DIM + c4 * 4;
        *(v4f*)(Xs + row * LDS_PITCH + c4 * 4) = *(const v4f*)gp;
        if (e0 + 128 < N_EDGES)                    // prefetch next tile
            __builtin_prefetch(gp + 128 * DIM, 0, 3);
    }
    __syncthreads();

    const int col = (lane & 15) + wave * 16;
    const int koff = (lane >> 4) * 2;
    const int arow0 = lane & 15;
    v2f bfrag[32];
#pragma unroll
    for (int kt = 0; kt < 32; ++kt) {
        int k = kt * 4 + koff;
        bfrag[kt].x = We[(size_t)k * DIM + col];
        bfrag[kt].y = We[(size_t)(k + 1) * DIM + col];
    }
    float bias = be[col];
#pragma unroll 1
    for (int m = 0; m < 8; ++m) {
        const float* xrow = Xs + (m * 16 + arow0) * LDS_PITCH + koff;
        v8f acc = {};
#pragma unroll
        for (int kt = 0; kt < 32; ++kt) {
            v2f a = *(const v2f*)(xrow + kt * 4);
            acc = __builtin_amdgcn_wmma_f32_16x16x4_f32(
                    false, a, false, bfrag[kt], (short)0, acc, false, false);
        }
        int mbase = m * 16 + (lane >> 4) * 8;
#pragma unroll
        for (int j = 0; j < 8; ++j)
            Xe[(mbase + j) * LDS_PITCH + col] = acc[j] + bias;
    }
    __syncthreads();
}

// -----------------------------------------------------------------------
// Kernel 2: attention logits a = <q[dst], k[src]+x_ij> * 1/sqrt(hd)
//           + atomic segment-max per (dst, head)
// -----------------------------------------------------------------------
__global__ void edge_logits_kernel(const float* __restrict__ e_ij,
                                   const int* __restrict__ src, const int* __restrict__ dst,
                                   const float* __restrict__ We, const float* __restrict__ be,
                                   const float* __restrict__ Q, const float* __restrict__ K,
                                   float* __restrict__ Aex, unsigned* __restrict__ amax) {
    extern __shared__ float lds[];
    float* Xs = lds;
    float* Xe = lds + 128 * LDS_PITCH;
    const int e0 = blockIdx.x * 128;
    edge_tile_gemm(e_ij, We, be, Xs, Xe, e0);

#pragma unroll
    for (int it = 0; it < 4; ++it) {               // 1024 (edge,head) pairs / 256 thr
        int p = threadIdx.x + 256 * it;
        int el = p >> 3, hh = p & 7;
        int e = e0 + el;
        int s = src[e], d = dst[e];
        const float* qp = Q + (size_t)d * DIM + hh * HEADDIM;
        const float* kp = K + (size_t)s * DIM + hh * HEADDIM;
        const float* xp = Xe + el * LDS_PITCH + hh * HEADDIM;
        float sum = 0.0f;
#pragma unroll
        for (int i = 0; i < HEADDIM; ++i)
            sum += qp[i] * (kp[i] + xp[i]);
        sum *= INV_SQRT_HD;
        Aex[(size_t)e * NHEAD + hh] = sum;
        atomicMax(&amax[d * NHEAD + hh], f32_ord(sum));
    }
}

// -----------------------------------------------------------------------
// Kernel 3: ex = exp(a - max[dst]); denom[dst] += ex (in-place on Aex)
// -----------------------------------------------------------------------
__global__ void softmax_kernel(const int* __restrict__ dst,
                               float* __restrict__ Aex,
                               const unsigned* __restrict__ amax,
                               float* __restrict__ denom) {
    int idx = blockIdx.x * 256 + threadIdx.x;
    if (idx >= N_EDGES * NHEAD) return;
    int e = idx >> 3, hh = idx & 7;
    int d = dst[e];
    float mx = ord_f32(amax[d * NHEAD + hh]);
    float ex = __expf(Aex[idx] - mx);
    Aex[idx] = ex;
    atomicAdd(&denom[d * NHEAD + hh], ex);
}

// -----------------------------------------------------------------------
// Kernel 4: recompute x_ij, m = attn*(v[src]+x_ij), atomic scatter to OUT
// -----------------------------------------------------------------------
__global__ void edge_agg_kernel(const float* __restrict__ e_ij,
                                const int* __restrict__ src, const int* __restrict__ dst,
                                const float* __restrict__ We, const float* __restrict__ be,
                                const float* __restrict__ V, const float* __restrict__ Aex,
                                const float* __restrict__ denom, float* __restrict__ OUT) {
    extern __shared__ float lds[];
    float* Xs = lds;
    float* Xe = lds + 128 * LDS_PITCH;
    const int e0 = blockIdx.x * 128;
    edge_tile_gemm(e_ij, We, be, Xs, Xe, e0);

#pragma unroll
    for (int it = 0; it < 4; ++it) {
        int p = threadIdx.x + 256 * it;
        int el = p >> 3, hh = p & 7;
        int e = e0 + el;
        int s = src[e], d = dst[e];
        float attn = Aex[(size_t)e * NHEAD + hh] / denom[d * NHEAD + hh];
        const float* vp = V + (size_t)s * DIM + hh * HEADDIM;
        const float* xp = Xe + el * LDS_PITCH + hh * HEADDIM;
        float* op = OUT + (size_t)d * DIM + hh * HEADDIM;
#pragma unroll
        for (int i = 0; i < HEADDIM; ++i)
            atomicAdd(&op[i], attn * (vp[i] + xp[i]));
    }
}

// -----------------------------------------------------------------------
extern "C" void kernel_launch(void* const* d_in, const int* in_sizes, int n_in,
                              void* d_out, int out_size, void* d_ws, size_t ws_size,
                              hipStream_t stream) {
    const float* h    = (const float*)d_in[0];
    const float* e_ij = (const float*)d_in[1];
    const int*   src  = (const int*)d_in[2];
    const int*   dst  = (const int*)d_in[3];
    const float* We   = (const float*)d_in[4];
    const float* be   = (const float*)d_in[5];
    const float* Wq   = (const float*)d_in[6];
    const float* bq   = (const float*)d_in[7];
    const float* Wk   = (const float*)d_in[8];
    const float* bk   = (const float*)d_in[9];
    const float* Wv   = (const float*)d_in[10];
    const float* bv   = (const float*)d_in[11];
    const float* Wh0  = (const float*)d_in[12];
    const float* bh0  = (const float*)d_in[13];
    float* OUT = (float*)d_out;

    // workspace layout (~106 MB)
    float* Q      = (float*)d_ws;
    float* K      = Q + (size_t)N_NODES * DIM;
    float* V      = K + (size_t)N_NODES * DIM;
    float* Aex    = V + (size_t)N_NODES * DIM;
    unsigned* amax = (unsigned*)(Aex + (size_t)N_EDGES * NHEAD);
    float* denom  = (float*)(amax + (size_t)N_NODES * NHEAD);

    const size_t lds1 = (size_t)128 * LDS_PITCH * sizeof(float);       // 67.6 KB
    const size_t lds2 = (size_t)2 * 128 * LDS_PITCH * sizeof(float);   // 135 KB

    hipLaunchKernelGGL(init_kernel,
                       dim3((N_NODES * NHEAD + 255) / 256), dim3(256), 0, stream,
                       amax, denom);
    hipLaunchKernelGGL(node_proj_kernel,
                       dim3((N_NODES + 127) / 128), dim3(256), lds1, stream,
                       h, Wq, bq, Wk, bk, Wv, bv, Wh0, bh0, Q, K, V, OUT);
    hipLaunchKernelGGL(edge_logits_kernel,
                       dim3(N_EDGES / 128), dim3(256), lds2, stream,
                       e_ij, src, dst, We, be, Q, K, Aex, amax);
    hipLaunchKernelGGL(softmax_kernel,
                       dim3((N_EDGES * NHEAD + 255) / 256), dim3(256), 0, stream,
                       dst, Aex, amax, denom);
    hipLaunchKernelGGL(edge_agg_kernel,
                       dim3(N_EDGES / 128), dim3(256), lds2, stream,
                       e_ij, src, dst, We, be, V, Aex, denom, OUT);
}